// AlignmentModule_27814208209592
// MI455X (gfx1250) — compile-verified
//
#include <hip/hip_runtime.h>
#include <math.h>

// ---------------------------------------------------------------------------
// AlignmentModule for MI455X (gfx1250, wave32, WMMA + TDM/async-LDS).
// Conv stacks + cross-distance GEMM on V_WMMA_F32_16X16X4_F32 (full fp32).
// tx tile staging via TENSOR_LOAD_TO_LDS (TDM, hardware LDS padding),
// ft/X staging via GLOBAL_LOAD_ASYNC_TO_LDS, tracked with ASYNC/TENSOR cnt.
// ---------------------------------------------------------------------------

typedef __attribute__((ext_vector_type(2))) float v2f;
typedef __attribute__((ext_vector_type(8))) float v8f;
typedef __attribute__((ext_vector_type(4))) unsigned int v4u;
typedef __attribute__((ext_vector_type(8))) unsigned int v8u;

#define B_  16
#define TT  512      // T_TEXT
#define TF  2048     // T_FEATS
#define AD  256      // adim
#define OD  80       // odim

__device__ __forceinline__ v8f wmma4(v2f a, v2f b, v8f c) {
  // D = A(16x4 f32) x B(4x16 f32) + C(16x16 f32)
  return __builtin_amdgcn_wmma_f32_16x16x4_f32(
      /*neg_a=*/false, a, /*neg_b=*/false, b,
      /*c_mod=*/(short)0, c, /*reuse_a=*/false, /*reuse_b=*/false);
}

// Group-segment-relative LDS byte offset (generic LDS addr = aperture:offset).
__device__ __forceinline__ unsigned int lds_off(const void* p) {
  return (unsigned int)(size_t)p;
}

// Async DMA: one b128 / b32 from global into LDS (ASYNCcnt tracked).
__device__ __forceinline__ void async_ld_b128(unsigned int dst, const void* src) {
  asm volatile("global_load_async_to_lds_b128 %0, %1, off"
               :: "v"(dst), "v"(src) : "memory");
}
__device__ __forceinline__ void async_ld_b32(unsigned int dst, const void* src) {
  asm volatile("global_load_async_to_lds_b32 %0, %1, off"
               :: "v"(dst), "v"(src) : "memory");
}
__device__ __forceinline__ void wait_async0() {
  asm volatile("s_wait_asynccnt 0x0" ::: "memory");
}

// TDM: 2D tile load (tile_d0 dwords x tile_d1 rows, row stride stride_d0
// dwords) into LDS with hardware padding of +4 dwords after every 16 dwords
// (=> LDS row stride 20 floats, bank-conflict-free). D# per ISA 08 §8.3/8.4.
__device__ __forceinline__ void tdm_load_2d(unsigned int ldsaddr, const void* gptr,
                                            unsigned int tensor_d0, unsigned int tensor_d1,
                                            unsigned int tile_d0, unsigned int tile_d1,
                                            unsigned int stride_d0) {
  unsigned long long ga = (unsigned long long)(size_t)gptr;
  v4u g0;
  g0[0] = 1u;                                    // count=1 valid user descriptor
  g0[1] = ldsaddr;                               // lds_addr
  g0[2] = (unsigned int)ga;                      // global_addr[31:0]
  g0[3] = (unsigned int)((ga >> 32) & 0x01FFFFFFu) | 0x80000000u; // [56:32]|type=2
  v8u g1;
  g1[0] = (2u << 16)      // data_size = 4B
        | (1u << 20)      // pad_enable
        | (3u << 22)      // pad_interval: 16 dwords
        | (3u << 25);     // pad_amount:   4 dwords
  g1[1] = (tensor_d0 & 0xFFFFu) << 16;                    // tensor_dim0[15:0]
  g1[2] = (tensor_d0 >> 16) | ((tensor_d1 & 0xFFFFu) << 16);
  g1[3] = (tensor_d1 >> 16) | (tile_d0 << 16);            // tile_dim0
  g1[4] = tile_d1;                                        // tile_dim1, tile_dim2=0
  g1[5] = stride_d0;                                      // dim0 stride low 32
  g1[6] = 0u;
  g1[7] = 0u;
  asm volatile("tensor_load_to_lds %0, %1" :: "s"(g0), "s"(g1) : "memory");
}

// ---------------------------------------------------------------------------
// Beta-binomial prior [TF, TT]; batch independent so computed once.
// betabinom_logpmf(k=t, n=512, a=f+1, b=2048-f); a+b = 2049 for every f.
// ---------------------------------------------------------------------------
__global__ void prior_kernel(float* __restrict__ prior) {
  int i = blockIdx.x * 256 + threadIdx.x;
  if (i >= TF * TT) return;
  int f = i / TT;
  int t = i - f * TT;
  float lp = lgammaf(513.0f)
           - lgammaf((float)(t + 1))
           - lgammaf((float)(513 - t))
           + lgammaf((float)(t + f + 1))
           + lgammaf((float)(2560 - t - f))
           - lgammaf(2561.0f)
           - lgammaf((float)(f + 1))
           - lgammaf((float)(2048 - f))
           + lgammaf(2049.0f);
  prior[i] = lp;
}

// ---------------------------------------------------------------------------
// Row-wise sum of squares over D (one wave32 per row).
// ---------------------------------------------------------------------------
__global__ void sumsq_kernel(const float* __restrict__ X, float* __restrict__ out,
                             int rows, int D) {
  int w = threadIdx.x >> 5, lane = threadIdx.x & 31;
  int row = blockIdx.x * 8 + w;
  if (row >= rows) return;
  const float* p = X + (size_t)row * D;
  float s = 0.0f;
  for (int i = lane; i < D; i += 32) { float v = p[i]; s += v * v; }
  #pragma unroll
  for (int off = 16; off; off >>= 1) s += __shfl_xor(s, off, 32);
  if (lane == 0) out[row] = s;
}

// ---------------------------------------------------------------------------
// Conv1d (k=1 or k=3, pad) as GEMM with WMMA f32 16x16x4.
// X: [B, T, CI] row-major, W: [CO=256, CI, KW] (torch), Y: [B, T, 256].
// Block tile: 128 rows x 32 out-channels; 8 waves, each wave = 16 rows x 32 ch.
// Im2col X staging uses GLOBAL_LOAD_ASYNC_TO_LDS_B32 (zeros via ds_store).
// ---------------------------------------------------------------------------
__global__ __launch_bounds__(256)
void conv_wmma_kernel(const float* __restrict__ X, const float* __restrict__ W,
                      const float* __restrict__ bias, float* __restrict__ Y,
                      int T, int CI, int KW, int pad, int relu) {
  const int KC = 32;
  __shared__ float Xs[128][KC + 2];   // stride 34 floats: bank-conflict-free
  __shared__ float Wt[32][KC + 2];    // W transposed to [n][k]

  int b   = blockIdx.z;
  int t0  = blockIdx.x * 128;
  int co0 = blockIdx.y * 32;
  int tid = threadIdx.x;
  int w = tid >> 5, lane = tid & 31;
  int hi = lane >> 4, lo = lane & 15;
  int m0 = w * 16;
  int K = CI * KW;
  unsigned int xs_base = lds_off(&Xs[0][0]);

  v8f acc0 = {0,0,0,0,0,0,0,0};
  v8f acc1 = {0,0,0,0,0,0,0,0};

  int nch = (K + KC - 1) / KC;
  for (int ch = 0; ch < nch; ++ch) {
    int kbase = ch * KC;
    // stage X tile via async-LDS DMA (im2col, k-order = dk*CI + ci)
    for (int i = tid; i < 128 * KC; i += 256) {
      int r = i >> 5, k = i & 31;
      int kidx = kbase + k;
      bool valid = false;
      size_t gofs = 0;
      if (kidx < K) {
        int dk = kidx / CI, ci = kidx - dk * CI;
        int t = t0 + r + dk - pad;
        if (t >= 0 && t < T) { valid = true; gofs = ((size_t)(b * T + t)) * CI + ci; }
      }
      if (valid) {
        async_ld_b32(xs_base + (unsigned int)((r * (KC + 2) + k) * 4), X + gofs);
      } else {
        Xs[r][k] = 0.0f;
      }
    }
    // stage W^T tile (tiny, strided gather -> plain loads)
    for (int i = tid; i < 32 * KC; i += 256) {
      int n = i >> 5, k = i & 31;
      int kidx = kbase + k;
      float v = 0.0f;
      if (kidx < K) {
        int dk = kidx / CI, ci = kidx - dk * CI;
        v = W[((size_t)(co0 + n) * CI + ci) * KW + dk];
      }
      Wt[n][k] = v;
    }
    wait_async0();
    __syncthreads();
    #pragma unroll
    for (int kk = 0; kk < KC / 4; ++kk) {
      // A frag: lanes 0-15 rows m, K = kk*4+{0,1}; lanes 16-31 same rows K+{2,3}
      v2f a  = *(const v2f*)&Xs[m0 + lo][kk * 4 + hi * 2];
      v2f b0 = *(const v2f*)&Wt[lo     ][kk * 4 + hi * 2];
      v2f b1 = *(const v2f*)&Wt[lo + 16][kk * 4 + hi * 2];
      acc0 = wmma4(a, b0, acc0);
      acc1 = wmma4(a, b1, acc1);
    }
    __syncthreads();
  }

  // Epilogue: C layout = vgpr r -> row m0+r (+8 for lanes 16-31), lane lo -> col
  float bv0 = bias[co0 + lo];
  float bv1 = bias[co0 + 16 + lo];
  #pragma unroll
  for (int r = 0; r < 8; ++r) {
    int row = t0 + m0 + r + hi * 8;
    float v0 = acc0[r] + bv0;
    float v1 = acc1[r] + bv1;
    if (relu) { v0 = fmaxf(v0, 0.0f); v1 = fmaxf(v1, 0.0f); }
    size_t base = ((size_t)(b * T + row)) * 256;
    Y[base + co0 + lo]      = v0;
    Y[base + co0 + 16 + lo] = v1;
  }
}

// ---------------------------------------------------------------------------
// Fused: cross = ft . tx^T (WMMA), dist = sqrt(f2 + t2 - 2 cross),
// log_softmax over t (block owns 32 f-rows x all 512 t-cols) + prior.
// tx chunk staged by the Tensor Data Mover (hardware-padded rows, stride 20),
// ft chunk by async b128 DMA. 8 waves = 2 (f) x 4 (t); 8 wmma tiles / wave.
// ---------------------------------------------------------------------------
__global__ __launch_bounds__(256)
void dist_softmax_kernel(const float* __restrict__ ft, const float* __restrict__ tx,
                         const float* __restrict__ f2, const float* __restrict__ t2,
                         const float* __restrict__ prior, float* __restrict__ out) {
  const int KC = 16;
  const int STR = 20;                 // LDS row stride (16 data + 4 pad dwords)
  __shared__ float Fts[32][STR];
  __shared__ float Txs[TT][STR];
  __shared__ float redmax[32][4];
  __shared__ float redsum[32][4];

  int b  = blockIdx.y;
  int f0 = blockIdx.x * 32;
  int tid = threadIdx.x;
  int w = tid >> 5, lane = tid & 31;
  int wm = w >> 2, wn = w & 3;
  int hi = lane >> 4, lo = lane & 15;
  unsigned int fts_base = lds_off(&Fts[0][0]);
  unsigned int txs_base = lds_off(&Txs[0][0]);

  v8f acc[8];
  #pragma unroll
  for (int j = 0; j < 8; ++j) acc[j] = (v8f){0,0,0,0,0,0,0,0};

  const float* ftb = ft + ((size_t)b * TF + f0) * AD;
  const float* txb = tx + ((size_t)b * TT) * AD;

  for (int ch = 0; ch < AD / KC; ++ch) {
    int kbase = ch * KC;
    // TDM: tx tile [512 x 16 dwords], tensor row stride 256 dwords.
    if (w == 0)
      tdm_load_2d(txs_base, txb + kbase, AD, TT, KC, TT, AD);
    // ft tile [32 x 16 dwords] via async b128 (4 per row, 16B-aligned both ends)
    if (tid < 128) {
      int r = tid >> 2, kq = (tid & 3) << 2;
      async_ld_b128(fts_base + (unsigned int)((r * STR + kq) * 4),
                    ftb + (size_t)r * AD + kbase + kq);
    }
    wait_async0();
    __builtin_amdgcn_s_wait_tensorcnt(0);
    __syncthreads();
    #pragma unroll
    for (int kk = 0; kk < KC / 4; ++kk) {
      v2f a = *(const v2f*)&Fts[wm * 16 + lo][kk * 4 + hi * 2];
      #pragma unroll
      for (int j = 0; j < 8; ++j) {
        v2f bf = *(const v2f*)&Txs[wn * 128 + j * 16 + lo][kk * 4 + hi * 2];
        acc[j] = wmma4(a, bf, acc[j]);
      }
    }
    __syncthreads();
  }

  // scores: -sqrt(max(f2 + t2 - 2*cross, 1e-12)) computed in-place in acc
  float f2v[8], t2v[8];
  #pragma unroll
  for (int r = 0; r < 8; ++r) f2v[r] = f2[(size_t)b * TF + f0 + wm * 16 + r + hi * 8];
  #pragma unroll
  for (int j = 0; j < 8; ++j) t2v[j] = t2[(size_t)b * TT + wn * 128 + j * 16 + lo];
  #pragma unroll
  for (int j = 0; j < 8; ++j)
    #pragma unroll
    for (int r = 0; r < 8; ++r) {
      float sq = f2v[r] + t2v[j] - 2.0f * acc[j][r];
      acc[j][r] = -sqrtf(fmaxf(sq, 1e-12f));
    }

  // row max: per-thread over j, shfl_xor within the 16-lane C-matrix half
  // (xor masks 1..8 never cross the hi/lo boundary), then cross-wave via LDS.
  float mx[8];
  #pragma unroll
  for (int r = 0; r < 8; ++r) {
    float m = acc[0][r];
    #pragma unroll
    for (int j = 1; j < 8; ++j) m = fmaxf(m, acc[j][r]);
    #pragma unroll
    for (int off = 8; off; off >>= 1) m = fmaxf(m, __shfl_xor(m, off, 32));
    mx[r] = m;
  }
  if (lo == 0) {
    #pragma unroll
    for (int r = 0; r < 8; ++r) redmax[wm * 16 + r + hi * 8][wn] = mx[r];
  }
  __syncthreads();
  float gm[8];
  #pragma unroll
  for (int r = 0; r < 8; ++r) {
    int row = wm * 16 + r + hi * 8;
    gm[r] = fmaxf(fmaxf(redmax[row][0], redmax[row][1]),
                  fmaxf(redmax[row][2], redmax[row][3]));
  }

  // sum of exp
  float sm[8];
  #pragma unroll
  for (int r = 0; r < 8; ++r) {
    float s = 0.0f;
    #pragma unroll
    for (int j = 0; j < 8; ++j) s += __expf(acc[j][r] - gm[r]);
    #pragma unroll
    for (int off = 8; off; off >>= 1) s += __shfl_xor(s, off, 32);
    sm[r] = s;
  }
  if (lo == 0) {
    #pragma unroll
    for (int r = 0; r < 8; ++r) redsum[wm * 16 + r + hi * 8][wn] = sm[r];
  }
  __syncthreads();
  float lz[8];
  #pragma unroll
  for (int r = 0; r < 8; ++r) {
    int row = wm * 16 + r + hi * 8;
    lz[r] = logf(redsum[row][0] + redsum[row][1] + redsum[row][2] + redsum[row][3]);
  }

  // write: log_softmax + prior
  #pragma unroll
  for (int j = 0; j < 8; ++j) {
    int t = wn * 128 + j * 16 + lo;
    #pragma unroll
    for (int r = 0; r < 8; ++r) {
      int f = f0 + wm * 16 + r + hi * 8;
      out[((size_t)b * TF + f) * TT + t] =
          acc[j][r] - gm[r] - lz[r] + prior[(size_t)f * TT + t];
    }
  }
}

// ---------------------------------------------------------------------------
extern "C" void kernel_launch(void* const* d_in, const int* in_sizes, int n_in,
                              void* d_out, int out_size, void* d_ws, size_t ws_size,
                              hipStream_t stream) {
  (void)in_sizes; (void)n_in; (void)out_size; (void)ws_size;
  const float* text  = (const float*)d_in[0];
  const float* feats = (const float*)d_in[1];
  const float* t_w1  = (const float*)d_in[2];
  const float* t_b1  = (const float*)d_in[3];
  const float* t_w2  = (const float*)d_in[4];
  const float* t_b2  = (const float*)d_in[5];
  const float* f_w1  = (const float*)d_in[6];
  const float* f_b1  = (const float*)d_in[7];
  const float* f_w2  = (const float*)d_in[8];
  const float* f_b2  = (const float*)d_in[9];
  const float* f_w3  = (const float*)d_in[10];
  const float* f_b3  = (const float*)d_in[11];

  float* ws    = (float*)d_ws;
  float* prior = ws;                        // TF*TT           = 1,048,576
  float* bufA  = prior + (size_t)TF * TT;   // B*TF*AD         = 8,388,608
  float* bufB  = bufA + (size_t)B_ * TF * AD;
  float* txTmp = bufB + (size_t)B_ * TF * AD;  // B*TT*AD      = 2,097,152
  float* txOut = txTmp + (size_t)B_ * TT * AD;
  float* t2    = txOut + (size_t)B_ * TT * AD; // B*TT
  float* f2    = t2 + (size_t)B_ * TT;         // B*TF

  prior_kernel<<<(TF * TT + 255) / 256, 256, 0, stream>>>(prior);

  // text: conv k3 + relu -> conv k1
  conv_wmma_kernel<<<dim3(TT / 128, 8, B_), 256, 0, stream>>>(
      text, t_w1, t_b1, txTmp, TT, AD, 3, 1, 1);
  conv_wmma_kernel<<<dim3(TT / 128, 8, B_), 256, 0, stream>>>(
      txTmp, t_w2, t_b2, txOut, TT, AD, 1, 0, 0);

  // feats: conv k3 + relu -> conv k3 + relu -> conv k1
  conv_wmma_kernel<<<dim3(TF / 128, 8, B_), 256, 0, stream>>>(
      feats, f_w1, f_b1, bufA, TF, OD, 3, 1, 1);
  conv_wmma_kernel<<<dim3(TF / 128, 8, B_), 256, 0, stream>>>(
      bufA, f_w2, f_b2, bufB, TF, AD, 3, 1, 1);
  conv_wmma_kernel<<<dim3(TF / 128, 8, B_), 256, 0, stream>>>(
      bufB, f_w3, f_b3, bufA, TF, AD, 1, 0, 0);

  sumsq_kernel<<<(B_ * TT) / 8, 256, 0, stream>>>(txOut, t2, B_ * TT, AD);
  sumsq_kernel<<<(B_ * TF) / 8, 256, 0, stream>>>(bufA, f2, B_ * TF, AD);

  dist_softmax_kernel<<<dim3(TF / 32, B_), 256, 0, stream>>>(
      bufA, txOut, f2, t2, prior, (float*)d_out);
}